// RejectionSampler_43267500540400
// MI455X (gfx1250) — compile-verified
//
#include <hip/hip_runtime.h>

#define B_    128
#define K_    8
#define V_    32000
#define BLOCK 1024
#define CHUNK 32              // elements per thread
#define NT    (V_ / CHUNK)    // 1000 active threads per row
#define TINY_F 1.17549435e-38f

// ---- CDNA5 async global->LDS path (probe via __has_builtin) ----------------
#if defined(__gfx1250__) && \
    __has_builtin(__builtin_amdgcn_global_load_async_to_lds_b128) && \
    __has_builtin(__builtin_amdgcn_s_wait_asynccnt)
#define HAVE_ASYNC 1
#else
#define HAVE_ASYNC 0
#endif

#if HAVE_ASYNC
// Builtin signature (from clang diagnostic): (v4i32 AS1*, v4i32 AS3*, imm, imm)
typedef int v4i_t __attribute__((vector_size(16)));
typedef __attribute__((address_space(1))) v4i_t g_v4i_t;
typedef __attribute__((address_space(3))) v4i_t l_v4i_t;
#endif

// ---------------------------------------------------------------------------
// Kernel 1: per (b,k) row — f = max(q-p,TINY), total S, inverse-CDF count.
// One workgroup per row, each thread owns 32 contiguous elements in regs.
// ---------------------------------------------------------------------------
__global__ __launch_bounds__(BLOCK)
void rs_recover_kernel(const float* __restrict__ target,
                       const float* __restrict__ draft,
                       const float* __restrict__ mult_u,
                       int* __restrict__ rec_ids) {
  const int row = blockIdx.x;       // 0 .. B_*K_-1
  const int t   = threadIdx.x;
  const bool active = (t < NT);

  __shared__ float s_sums[BLOCK];
  __shared__ float s_total;
  __shared__ int   s_cnt;
#if HAVE_ASYNC
  __shared__ float4 s_stg_t[2][BLOCK];
  __shared__ float4 s_stg_d[2][BLOCK];
#endif
  if (t == 0) s_cnt = 0;

  const size_t base = (size_t)row * (size_t)V_ + (size_t)t * CHUNK;
  const float4* tg = (const float4*)(target + base);
  const float4* dg = (const float4*)(draft  + base);

  float4 fr[8];
  float  lsum = 0.0f;

#if HAVE_ASYNC
  // Double-buffered async DMA: global -> LDS stage, tracked by ASYNCcnt.
  if (active) {
    __builtin_amdgcn_global_load_async_to_lds_b128(
        (g_v4i_t*)(tg + 0), (l_v4i_t*)&s_stg_t[0][t], 0, 0);
    __builtin_amdgcn_global_load_async_to_lds_b128(
        (g_v4i_t*)(dg + 0), (l_v4i_t*)&s_stg_d[0][t], 0, 0);
  }
#pragma unroll
  for (int j = 0; j < 7; ++j) {
    if (active) {
      __builtin_amdgcn_global_load_async_to_lds_b128(
          (g_v4i_t*)(tg + j + 1), (l_v4i_t*)&s_stg_t[(j + 1) & 1][t], 0, 0);
      __builtin_amdgcn_global_load_async_to_lds_b128(
          (g_v4i_t*)(dg + j + 1), (l_v4i_t*)&s_stg_d[(j + 1) & 1][t], 0, 0);
    }
    __builtin_amdgcn_s_wait_asynccnt(2);   // iteration j's two loads landed
    float4 tv = s_stg_t[j & 1][t];
    float4 dv = s_stg_d[j & 1][t];
    float4 f;
    f.x = fmaxf(tv.x - dv.x, TINY_F);
    f.y = fmaxf(tv.y - dv.y, TINY_F);
    f.z = fmaxf(tv.z - dv.z, TINY_F);
    f.w = fmaxf(tv.w - dv.w, TINY_F);
    fr[j] = f;
    lsum += (f.x + f.y) + (f.z + f.w);
  }
  __builtin_amdgcn_s_wait_asynccnt(0);
  {
    float4 tv = s_stg_t[7 & 1][t];
    float4 dv = s_stg_d[7 & 1][t];
    float4 f;
    f.x = fmaxf(tv.x - dv.x, TINY_F);
    f.y = fmaxf(tv.y - dv.y, TINY_F);
    f.z = fmaxf(tv.z - dv.z, TINY_F);
    f.w = fmaxf(tv.w - dv.w, TINY_F);
    fr[7] = f;
    if (active) lsum += (f.x + f.y) + (f.z + f.w);
  }
  if (!active) lsum = 0.0f;
#else
  if (active) {
#pragma unroll
    for (int j = 0; j < 8; ++j) {
      float4 tv = tg[j];
      float4 dv = dg[j];
      float4 f;
      f.x = fmaxf(tv.x - dv.x, TINY_F);
      f.y = fmaxf(tv.y - dv.y, TINY_F);
      f.z = fmaxf(tv.z - dv.z, TINY_F);
      f.w = fmaxf(tv.w - dv.w, TINY_F);
      fr[j] = f;
      lsum += (f.x + f.y) + (f.z + f.w);
    }
  } else {
#pragma unroll
    for (int j = 0; j < 8; ++j) fr[j] = make_float4(0.f, 0.f, 0.f, 0.f);
  }
#endif

  s_sums[t] = lsum;
  __syncthreads();

  // Block-wide exclusive scan of the 1024 chunk sums, done by wave 0.
  if (t < 32) {
    float run = 0.0f;
    for (int j = 0; j < 32; ++j) {
      float v = s_sums[t * 32 + j];
      s_sums[t * 32 + j] = run;     // exclusive within group
      run += v;
    }
    float incl = run;
#pragma unroll
    for (int off = 1; off < 32; off <<= 1) {
      float n = __shfl_up(incl, off, 32);
      if (t >= off) incl += n;
    }
    float grp_base = incl - run;    // exclusive across groups
    for (int j = 0; j < 32; ++j) s_sums[t * 32 + j] += grp_base;
    if (t == 31) s_total = incl;    // grand total S
  }
  __syncthreads();

  const float u = mult_u[row] * s_total;

  int cnt = 0;
  if (active) {
    float run = s_sums[t];          // exclusive prefix at chunk start
#pragma unroll
    for (int j = 0; j < 8; ++j) {
      run += fr[j].x; cnt += (run < u) ? 1 : 0;
      run += fr[j].y; cnt += (run < u) ? 1 : 0;
      run += fr[j].z; cnt += (run < u) ? 1 : 0;
      run += fr[j].w; cnt += (run < u) ? 1 : 0;
    }
  }
#pragma unroll
  for (int off = 16; off > 0; off >>= 1) cnt += __shfl_down(cnt, off, 32);
  if ((t & 31) == 0 && cnt != 0) atomicAdd(&s_cnt, cnt);
  __syncthreads();

  if (t == 0) rec_ids[row] = min(s_cnt, V_ - 1);
}

// ---------------------------------------------------------------------------
// Kernel 2: accept test + output assembly. One thread per batch row.
// ---------------------------------------------------------------------------
__global__ void rs_finalize_kernel(const float* __restrict__ target,
                                   const float* __restrict__ draft,
                                   const float* __restrict__ uniform_rand,
                                   const int* __restrict__ bonus_ids,
                                   const int* __restrict__ draft_ids,
                                   const int* __restrict__ rec_ids,
                                   int* __restrict__ out) {
  const int b = blockIdx.x * blockDim.x + threadIdx.x;
  if (b >= B_) return;

  int limit = K_;
#pragma unroll
  for (int k = 0; k < K_; ++k) {
    const int    id  = draft_ids[b * K_ + k];
    const size_t off = (size_t)(b * K_ + k) * (size_t)V_ + (size_t)id;
    const float  p   = draft[off];
    const float  q   = target[off];
    const float  ratio = fminf(q / p, 1.0f);
    const bool   acc   = uniform_rand[b * K_ + k] < ratio;
    if (!acc && limit == K_) limit = k;
  }

#pragma unroll
  for (int k = 0; k < K_; ++k) {
    int v;
    if (k < limit)       v = draft_ids[b * K_ + k];
    else if (k == limit) v = rec_ids[b * K_ + k];
    else                 v = -1;
    out[b * (K_ + 1) + k] = v;
  }
  // bonus iff output[:, K-1] != -1  <=>  limit >= K-1
  out[b * (K_ + 1) + K_] = (limit >= K_ - 1) ? bonus_ids[b] : -1;
}

// ---------------------------------------------------------------------------
extern "C" void kernel_launch(void* const* d_in, const int* in_sizes, int n_in,
                              void* d_out, int out_size, void* d_ws, size_t ws_size,
                              hipStream_t stream) {
  (void)in_sizes; (void)n_in; (void)out_size; (void)ws_size;
  const float* target       = (const float*)d_in[0];
  const float* draft        = (const float*)d_in[1];
  const float* uniform_rand = (const float*)d_in[2];
  const float* mult_u       = (const float*)d_in[3];
  const int*   bonus_ids    = (const int*)d_in[4];
  const int*   draft_ids    = (const int*)d_in[5];
  int* out = (int*)d_out;
  int* rec = (int*)d_ws;    // B_*K_ recovered token ids

  rs_recover_kernel<<<dim3(B_ * K_), dim3(BLOCK), 0, stream>>>(target, draft, mult_u, rec);
  rs_finalize_kernel<<<dim3(1), dim3(128), 0, stream>>>(target, draft, uniform_rand,
                                                        bonus_ids, draft_ids, rec, out);
}